// PointPillarsScatter_2903397892133
// MI455X (gfx1250) — compile-verified
//
#include <hip/hip_runtime.h>

// PointPillarsScatter for MI455X (gfx1250).
// Pure bandwidth-bound scatter: ~110 MB zero-fill + 6 MB coalesced read +
// 6 MB scattered write. Zero FLOPs -> WMMA is not applicable; the roofline is
// the 23.3 TB/s store path (~5 us floor), so optimize vmem width/issue count.

// Native clang vector type: required by __builtin_nontemporal_load (HIP's
// float4 is a class and is rejected), lowers to global_load_b128 th:NT.
typedef float v4f __attribute__((ext_vector_type(4)));

namespace {
constexpr int kNY   = 496;
constexpr int kNX   = 432;
constexpr int kC    = 64;
constexpr int kB    = 2;
constexpr int kNPer = 12000;
constexpr int kNTot = kB * kNPer;                 // 24000 voxels
constexpr int kPlane = kNY * kNX;                 // 214272 elements per (b,c) plane
constexpr int kOutElems = kB * kC * kPlane;       // 27,426,816 floats (~110 MB)
constexpr int kOutVec4  = kOutElems / 4;          // 6,856,704 (kNX % 4 == 0)

constexpr int kFillUnroll  = 4;                   // 4 x b128 stores per thread
constexpr int kFillThreads = 256;
// 6,856,704 / 1024 = 6696 exactly -> no tail, no bounds checks needed.
static_assert(kOutVec4 % (kFillThreads * kFillUnroll) == 0, "fill tail");
constexpr int kFillBlocks = kOutVec4 / (kFillThreads * kFillUnroll);

constexpr int kScatThreads = 256;
// 24000 * 16 = 384,000 = 1500 * 256 exactly -> no tail.
static_assert((kNTot * (kC / 4)) % kScatThreads == 0, "scatter tail");
constexpr int kScatBlocks = (kNTot * (kC / 4)) / kScatThreads;
}  // namespace

// Kernel 1: zero the canvas with four unconditional 128-bit stores per thread
// (global_store_b128 with folded immediate offsets). Default RT temporal hint
// keeps the canvas resident in the 192 MB L2 so the subsequent scattered 4B
// stores hit L2 lines instead of HBM.
__global__ void pps_zero_fill(v4f* __restrict__ out) {
    const v4f z = {0.0f, 0.0f, 0.0f, 0.0f};
    // Block-contiguous layout: block i owns 1024 consecutive float4s; each
    // thread's 4 stores are 4 KB apart -> every store of a wave32 is a
    // contiguous 512 B burst.
    int i = blockIdx.x * (kFillThreads * kFillUnroll) + threadIdx.x;
#pragma unroll
    for (int u = 0; u < kFillUnroll; ++u) {
        out[i] = z;
        i += kFillThreads;
    }
}

// Kernel 2: scatter voxel features into the canvas.
// Thread t handles voxel n = t/16, channel quad q = t%16 (channels 4q..4q+3):
//   - feature read: one v4f (global_load_b128 th:NT), fully coalesced
//     (a wave32 reads 512 contiguous bytes of the feature matrix)
//   - coords read once per 4 channels
//   - writes: 4 plane-strided b32 stores -> the irreducible 857 KB-stride
//     scatter, absorbed by the L2-resident canvas.
// Streamed read-once data uses non-temporal loads (gfx1250 TH=NT) to avoid
// displacing canvas lines from the near caches.
__global__ void pps_scatter(const v4f* __restrict__ feats4,
                            const int* __restrict__ coords,
                            float* __restrict__ out) {
    const int t = blockIdx.x * blockDim.x + threadIdx.x;
    const int n = t >> 4;           // voxel index (< kNTot by exact launch)
    const int q = t & 15;           // channel quad: channels 4q .. 4q+3

    const int bid = __builtin_nontemporal_load(&coords[n * 4 + 0]);
    const int y   = __builtin_nontemporal_load(&coords[n * 4 + 2]);
    const int x   = __builtin_nontemporal_load(&coords[n * 4 + 3]);

    const v4f v = __builtin_nontemporal_load(&feats4[n * (kC / 4) + q]);

    const int lin = y * kNX + x;                            // < 214272
    const int c0  = q * 4;
    int idx = (bid * kC + c0) * kPlane + lin;               // < 27.43M, int32 ok

    out[idx] = v.x;  idx += kPlane;
    out[idx] = v.y;  idx += kPlane;
    out[idx] = v.z;  idx += kPlane;
    out[idx] = v.w;
}

extern "C" void kernel_launch(void* const* d_in, const int* in_sizes, int n_in,
                              void* d_out, int out_size, void* d_ws, size_t ws_size,
                              hipStream_t stream) {
    (void)in_sizes; (void)n_in; (void)out_size; (void)d_ws; (void)ws_size;

    const float* feats  = (const float*)d_in[0];   // (24000, 64) fp32
    const int*   coords = (const int*)d_in[1];     // (24000, 4) int
    float*       out    = (float*)d_out;           // (2, 64, 496, 432) fp32

    // Zero-fill: 6,856,704 b128 stores, 4 per thread, exact cover.
    pps_zero_fill<<<kFillBlocks, kFillThreads, 0, stream>>>((v4f*)out);

    // Scatter: 24000 voxels * 16 channel-quads = 384,000 threads, exact cover.
    pps_scatter<<<kScatBlocks, kScatThreads, 0, stream>>>(
        (const v4f*)feats, coords, out);
}